// GraphAttentionLayer_57363583205947
// MI455X (gfx1250) — compile-verified
//
#include <hip/hip_runtime.h>
#include <math.h>

// Types for WMMA fragments (wave32: 16x4 f32 A = 2 floats/lane, 16x16 f32 C = 8 floats/lane)
typedef float v2f __attribute__((ext_vector_type(2)));
typedef float v8f __attribute__((ext_vector_type(8)));

#define FDIM 128
#define LEAKY 0.2f

// ---------------------------------------------------------------------------
// Kernel 1: w1 = W_map @ a1, w2 = W_map @ a2   (128x128 @ 128 -> 128, tiny)
// ---------------------------------------------------------------------------
__global__ void gat_wprep(const float* __restrict__ W_map,
                          const float* __restrict__ a1,
                          const float* __restrict__ a2,
                          float* __restrict__ w1,
                          float* __restrict__ w2) {
    int f = threadIdx.x;           // 0..127
    if (f >= FDIM) return;
    const float* row = W_map + (size_t)f * FDIM;
    float acc1 = 0.f, acc2 = 0.f;
    #pragma unroll 8
    for (int d = 0; d < FDIM; ++d) {
        float w = row[d];
        acc1 += w * a1[d];
        acc2 += w * a2[d];
    }
    w1[f] = acc1;
    w2[f] = acc2;
}

// ---------------------------------------------------------------------------
// Kernel 2: row_ptr[i] = lower_bound(edge_row, i), edge_row is sorted
// ---------------------------------------------------------------------------
__global__ void gat_rowptr(const int* __restrict__ edge_row,
                           int* __restrict__ row_ptr, int N, int E) {
    int i = blockIdx.x * blockDim.x + threadIdx.x;
    if (i > N) return;
    int lo = 0, hi = E;
    while (lo < hi) {
        int mid = (lo + hi) >> 1;
        if (edge_row[mid] < i) lo = mid + 1; else hi = mid;
    }
    row_ptr[i] = lo;
}

// ---------------------------------------------------------------------------
// Kernel 3: s1[i] = dot(x[i], w1) + b1 ; s2[i] = dot(x[i], w2) + b2
// One wave32 per row; each lane owns 4 contiguous floats (float4 load).
// ---------------------------------------------------------------------------
__global__ void gat_scores(const float* __restrict__ x,
                           const float* __restrict__ w1,
                           const float* __restrict__ w2,
                           const float* __restrict__ b1p,
                           const float* __restrict__ b2p,
                           float* __restrict__ s1,
                           float* __restrict__ s2, int N) {
    const int lane = threadIdx.x & 31;
    const int row  = blockIdx.x * (blockDim.x >> 5) + (threadIdx.x >> 5);
    if (row >= N) return;
    const float4 xv  = *(const float4*)(x  + (size_t)row * FDIM + lane * 4);
    const float4 w1v = ((const float4*)w1)[lane];
    const float4 w2v = ((const float4*)w2)[lane];
    float d1 = xv.x * w1v.x + xv.y * w1v.y + xv.z * w1v.z + xv.w * w1v.w;
    float d2 = xv.x * w2v.x + xv.y * w2v.y + xv.z * w2v.z + xv.w * w2v.w;
    #pragma unroll
    for (int off = 16; off > 0; off >>= 1) {
        d1 += __shfl_xor(d1, off, 32);
        d2 += __shfl_xor(d2, off, 32);
    }
    if (lane == 0) {
        s1[row] = d1 + b1p[0];
        s2[row] = d2 + b2p[0];
    }
}

// ---------------------------------------------------------------------------
// Kernel 4: value = x @ kernel via V_WMMA_F32_16X16X4_F32 (exact fp32).
// Block = 256 threads = 8 waves. Weight matrix staged in LDS in K-pair
// interleaved order: ldsW[(k>>1)*256 + col*2 + (k&1)], so each lane's B
// fragment (K=kbase, K=kbase+1 at one column) is ONE aligned ds_load_b64
// straight into an even VGPR pair — no repack movs before the WMMA.
// Each wave computes a 16-row x 128-col output strip: 32 K-steps x 8 N-tiles.
//
// ISA layouts (CDNA5 ISA §7.12.2):
//   A 16x4 f32 : lanes 0-15 M=0..15 (VGPR0 K=0, VGPR1 K=1); lanes 16-31 K=2,3
//   B 4x16 f32 : lanes hold N=lane&15; lane-half selects K pair (mirrors A)
//   C/D 16x16  : VGPR v -> M=v (lanes 0-15) / M=v+8 (lanes 16-31), N=lane&15
// ---------------------------------------------------------------------------
__global__ void gat_value_gemm(const float* __restrict__ x,
                               const float* __restrict__ Wk,
                               float* __restrict__ value, int N) {
    __shared__ float ldsW[FDIM * FDIM];     // 64 KB of 320 KB/WGP

    // Cooperative LDS fill, interleaving K pairs (coalesced global reads).
    for (int i = threadIdx.x; i < FDIM * FDIM; i += blockDim.x) {
        const int k = i >> 7;       // K row 0..127
        const int c = i & 127;      // column 0..127
        ldsW[((k >> 1) << 8) + (c << 1) + (k & 1)] = Wk[i];
    }
    __syncthreads();

    const int lane   = threadIdx.x & 31;
    const int l16    = lane & 15;
    const int lhalf  = lane >> 4;            // 0 or 1
    const int waveId = blockIdx.x * (blockDim.x >> 5) + (threadIdx.x >> 5);
    const int row0   = waveId * 16;
    if (row0 >= N) return;                   // uniform per wave -> EXEC all-1s

    v8f acc[8] = {};                         // 8 n-tiles x 8 VGPRs = 16x128 f32

    int arow = row0 + l16;                   // A row owned by this lane
    if (arow >= N) arow = N - 1;             // clamp (partial strip; safe)
    const float* xrow = x + (size_t)arow * FDIM;

    for (int kk = 0; kk < 32; ++kk) {
        const int kbase = kk * 4 + lhalf * 2;
        const v2f a = *(const v2f*)(xrow + kbase);          // K = kbase, kbase+1
        // One K-pair row of interleaved LDS: pair index = kbase>>1 = kk*2+lhalf
        const float* ldsRow = ldsW + (((kk << 1) + lhalf) << 8);
        #pragma unroll
        for (int n = 0; n < 8; ++n) {
            const int ncol = n * 16 + l16;
            const v2f b = *(const v2f*)(ldsRow + (ncol << 1));   // ds_load_b64
            acc[n] = __builtin_amdgcn_wmma_f32_16x16x4_f32(
                false, a, false, b, (short)0, acc[n], false, false);
        }
    }

    // Store D: VGPR v -> output row row0 + v + lhalf*8, col n*16 + l16
    #pragma unroll
    for (int n = 0; n < 8; ++n) {
        #pragma unroll
        for (int v = 0; v < 8; ++v) {
            const int m = v + lhalf * 8;
            if (row0 + m < N)
                value[(size_t)(row0 + m) * FDIM + n * 16 + l16] = acc[n][v];
        }
    }
}

// ---------------------------------------------------------------------------
// Kernel 5: per-row segment softmax + SpMM + bias.  One wave32 per row.
//   pass1: m   = max_e leaky(ev*s1[r] + ev*s2[col])     (lanes stride edges)
//   pass2: den = sum_e exp(e - m)
//   pass3: out[r,:] = sum_e (exp(e-m)/den) * value[col,:] + bias[r,:]
//          (each lane owns 4 contiguous cols -> float4 L2-resident gather)
// ---------------------------------------------------------------------------
__global__ void gat_edge_spmm(const int*   __restrict__ edge_col,
                              const float* __restrict__ edge_val,
                              const float* __restrict__ s1,
                              const float* __restrict__ s2,
                              const int*   __restrict__ row_ptr,
                              const float* __restrict__ value,
                              const float* __restrict__ bias,
                              float* __restrict__ out, int N) {
    const int lane = threadIdx.x & 31;
    const int row  = blockIdx.x * (blockDim.x >> 5) + (threadIdx.x >> 5);
    if (row >= N) return;

    const int start = row_ptr[row];
    const int end   = row_ptr[row + 1];
    const float sr1 = s1[row];

    // ---- pass 1: segment max of leaky-relu'd scores
    float m = -INFINITY;
    for (int e = start + lane; e < end; e += 32) {
        const float ev = edge_val[e];
        float sc = ev * sr1 + ev * s2[edge_col[e]];
        sc = sc > 0.f ? sc : LEAKY * sc;
        m = fmaxf(m, sc);
    }
    #pragma unroll
    for (int off = 16; off > 0; off >>= 1)
        m = fmaxf(m, __shfl_xor(m, off, 32));
    if (!(m > -INFINITY)) m = 0.f;           // empty row -> m := 0

    // ---- pass 2: denominator
    float den = 0.f;
    for (int e = start + lane; e < end; e += 32) {
        const float ev = edge_val[e];
        float sc = ev * sr1 + ev * s2[edge_col[e]];
        sc = sc > 0.f ? sc : LEAKY * sc;
        den += __expf(sc - m);
    }
    #pragma unroll
    for (int off = 16; off > 0; off >>= 1)
        den += __shfl_xor(den, off, 32);
    const float inv_den = (den > 0.f) ? (1.f / den) : 0.f;

    // ---- pass 3: weighted accumulation of value rows (serial over edges,
    //      512B per edge across the wave; value fits in L2)
    float4 acc = make_float4(0.f, 0.f, 0.f, 0.f);
    const int c4 = lane * 4;
    for (int e = start; e < end; ++e) {
        const int   c  = edge_col[e];
        const float ev = edge_val[e];
        float sc = ev * sr1 + ev * s2[c];
        sc = sc > 0.f ? sc : LEAKY * sc;
        const float att = __expf(sc - m) * inv_den;
        const float4 v4 = *(const float4*)(value + (size_t)c * FDIM + c4);
        acc.x += att * v4.x;
        acc.y += att * v4.y;
        acc.z += att * v4.z;
        acc.w += att * v4.w;
    }
    const float4 b4 = *(const float4*)(bias + (size_t)row * FDIM + c4);
    float4 o;
    o.x = acc.x + b4.x; o.y = acc.y + b4.y;
    o.z = acc.z + b4.z; o.w = acc.w + b4.w;
    *(float4*)(out + (size_t)row * FDIM + c4) = o;
}

// ---------------------------------------------------------------------------
// Launch
// ---------------------------------------------------------------------------
static inline size_t align256(size_t v) { return (v + 255) & ~(size_t)255; }

extern "C" void kernel_launch(void* const* d_in, const int* in_sizes, int n_in,
                              void* d_out, int out_size, void* d_ws, size_t ws_size,
                              hipStream_t stream) {
    const float* x        = (const float*)d_in[0];
    const int*   edge_row = (const int*)  d_in[1];
    const int*   edge_col = (const int*)  d_in[2];
    const float* edge_val = (const float*)d_in[3];
    const float* W_map    = (const float*)d_in[4];
    const float* a1       = (const float*)d_in[5];
    const float* b1p      = (const float*)d_in[6];
    const float* a2       = (const float*)d_in[7];
    const float* b2p      = (const float*)d_in[8];
    const float* Wk       = (const float*)d_in[9];
    const float* bias     = (const float*)d_in[10];
    float*       out      = (float*)d_out;

    const int N = in_sizes[0] / FDIM;   // x is [N, 128]
    const int E = in_sizes[1];          // edge_row length

    // Workspace layout
    char* ws = (char*)d_ws;
    size_t off = 0;
    float* value = (float*)(ws + off);  off = align256(off + (size_t)N * FDIM * sizeof(float));
    float* s1    = (float*)(ws + off);  off = align256(off + (size_t)N * sizeof(float));
    float* s2    = (float*)(ws + off);  off = align256(off + (size_t)N * sizeof(float));
    float* w1    = (float*)(ws + off);  off = align256(off + FDIM * sizeof(float));
    float* w2    = (float*)(ws + off);  off = align256(off + FDIM * sizeof(float));
    int*   rptr  = (int*)  (ws + off);  off = align256(off + (size_t)(N + 1) * sizeof(int));
    (void)ws_size;

    // 1) fold attention vectors through W_map
    gat_wprep<<<1, FDIM, 0, stream>>>(W_map, a1, a2, w1, w2);

    // 2) CSR row pointers from sorted edge_row
    {
        int threads = 256, blocks = (N + 1 + threads - 1) / threads;
        gat_rowptr<<<blocks, threads, 0, stream>>>(edge_row, rptr, N, E);
    }

    // 3) per-node scores s1, s2  (wave per row, 8 waves/block)
    {
        int blocks = (N + 7) / 8;
        gat_scores<<<blocks, 256, 0, stream>>>(x, w1, w2, b1p, b2p, s1, s2, N);
    }

    // 4) value = x @ kernel via fp32 WMMA (wave per 16-row strip, 8 waves/block)
    {
        int waves  = (N + 15) / 16;
        int blocks = (waves + 7) / 8;
        gat_value_gemm<<<blocks, 256, 0, stream>>>(x, Wk, value, N);
    }

    // 5) segment softmax + SpMM + bias (wave per row)
    {
        int blocks = (N + 7) / 8;
        gat_edge_spmm<<<blocks, 256, 0, stream>>>(edge_col, edge_val, s1, s2,
                                                  rptr, value, bias, out, N);
    }
}